// VectoriSHT_75660143886826
// MI455X (gfx1250) — compile-verified
//
#include <hip/hip_runtime.h>
#include <hip/hip_bf16.h>

typedef __attribute__((ext_vector_type(2))) float v2f;
typedef __attribute__((ext_vector_type(8))) float v8f;

#define NLAT 511
#define LMAX 511
#define MMAX 257

// Grid: (MMAX, 16). Block: 128 threads = 4 wave32.
// Wave w: d_idx = w&1, k-subtile = w>>1. Each wave accumulates a 32(M) x 16(K)
// tile of P_d = [xr0;xi0;xr1;xi1](32 x L) @ d[d_idx][m](L x K) with
// v_wmma_f32_16x16x4_f32. dpct is streamed once from HBM (non-temporal,
// software-pipelined + prefetched); x panels live in LDS (and L2 across blocks).
__global__ __launch_bounds__(128) void sht_wmma_kernel(const float* __restrict__ x,
                                                       const float* __restrict__ dpct,
                                                       float* __restrict__ out)
{
    // A panel, K-pair layout: element (row, l) at [(l>>1)*64 + row*2 + (l&1)]
    __shared__ float ldsA[512 * 32];              // 64 KB
    // staging: [ksub][d][k(16)][row(32)] -> v8f-contiguous per accumulator
    __shared__ float ldsC[2 * 2 * 16 * 32];       // 8 KB

    const int m   = blockIdx.x;   // 0..256
    const int kt2 = blockIdx.y;   // 0..15 (32 k values per block)
    const int tid = threadIdx.x;
    const int l0  = m & ~3;       // triangular mask: d[m,l,k]==0 for l<m

    // ---- Stage A panel into LDS: row = (c*2+ri)*8 + b, value x[b,c,l,m,ri] ----
    {
        const int n = (512 - l0) * 32;
        for (int i = tid; i < n; i += 128) {
            int l   = l0 + (i >> 5);
            int row = i & 31;
            float v = 0.0f;
            if (l < LMAX) {
                int b   = row & 7;
                int cri = row >> 3;          // 0:xr0 1:xi0 2:xr1 3:xi1
                int c   = cri >> 1;
                int ri  = cri & 1;
                v = x[((((size_t)b * 2 + c) * LMAX + l) * MMAX + m) * 2 + ri];
            }
            ldsA[(l >> 1) * 64 + row * 2 + (l & 1)] = v;  // l==511 slot zeroed
        }
    }
    __syncthreads();

    const int w     = tid >> 5;
    const int lane  = tid & 31;
    const int d_idx = w & 1;
    const int ksub  = w >> 1;
    const int nlo   = lane & 15;     // N column / A-row within tile
    const int lh    = lane >> 4;     // K-pair selector (0: K=0,1  1: K=2,3)
    const int kcol  = kt2 * 32 + ksub * 16 + nlo;
    const int kcl   = (kcol < NLAT) ? kcol : (NLAT - 1); // clamped: k==511 column
                                                          // computed but discarded
    const float* dm = dpct + ((size_t)d_idx * MMAX + m) * (size_t)LMAX * NLAT;
    const float* pB = dm + (size_t)(l0 + 2 * lh) * NLAT + kcl;

    v8f accLo = {};   // M rows 0..15
    v8f accHi = {};   // M rows 16..31

    // ---- Software-pipelined main loop (all L accesses in-bounds for l<=507) ----
    int l = l0;
    // prologue: load B for l0
    float bx = __builtin_nontemporal_load(pB);
    float by = __builtin_nontemporal_load(pB + NLAT);
    pB += (size_t)4 * NLAT;           // now points at rows for l+4

    #pragma unroll 4
    for (; l < LMAX - 3; l += 4) {
        // Issue NEXT iteration's B loads first (uniform condition: l is scalar)
        float nbx = 0.0f, nby = 0.0f;
        if (l + 4 < LMAX - 3) {
            nbx = __builtin_nontemporal_load(pB);          // d[m, l+4+2lh,   kcl]
            nby = __builtin_nontemporal_load(pB + NLAT);   // d[m, l+4+2lh+1, kcl]
        }
        if (l < LMAX - 35)                                  // uniform scalar guard
            __builtin_prefetch(pB + (size_t)28 * NLAT, 0, 3);

        // Consume CURRENT iteration's values
        v2f bf = {bx, by};
        const int pa = (l >> 1) + lh;                       // K-pair index
        v2f aLo = *reinterpret_cast<const v2f*>(&ldsA[pa * 64 + nlo * 2]);
        v2f aHi = *reinterpret_cast<const v2f*>(&ldsA[pa * 64 + 32 + nlo * 2]);

        accLo = __builtin_amdgcn_wmma_f32_16x16x4_f32(false, aLo, false, bf,
                                                      (short)0, accLo, false, false);
        accHi = __builtin_amdgcn_wmma_f32_16x16x4_f32(false, aHi, false, bf,
                                                      (short)0, accHi, false, false);

        bx = nbx; by = nby;
        pB += (size_t)4 * NLAT;
    }

    // ---- Peeled tail (l == 508): row l+2lh+1 may be 511 -> guard ----
    {
        // pB advanced one step past: rewind to tail rows
        const float* pT = pB - (size_t)4 * NLAT;
        float tbx = __builtin_nontemporal_load(pT);         // l+2lh <= 510: valid
        float tby = 0.0f;
        if (lh == 0) tby = __builtin_nontemporal_load(pT + NLAT);
        v2f bf = {tbx, tby};
        const int pa = (l >> 1) + lh;
        v2f aLo = *reinterpret_cast<const v2f*>(&ldsA[pa * 64 + nlo * 2]);
        v2f aHi = *reinterpret_cast<const v2f*>(&ldsA[pa * 64 + 32 + nlo * 2]);
        accLo = __builtin_amdgcn_wmma_f32_16x16x4_f32(false, aLo, false, bf,
                                                      (short)0, accLo, false, false);
        accHi = __builtin_amdgcn_wmma_f32_16x16x4_f32(false, aHi, false, bf,
                                                      (short)0, accHi, false, false);
    }

    // ---- Stage accumulators: contiguous v8f -> 2x ds_store_b128 each ----
    {
        float* cptr = &ldsC[(((ksub * 2 + d_idx) * 16 + nlo) * 32) + 8 * lh];
        *reinterpret_cast<v8f*>(cptr)      = accLo;   // rows 8lh .. 8lh+7
        *reinterpret_cast<v8f*>(cptr + 16) = accHi;   // rows 16+8lh .. 16+8lh+7
    }
    __syncthreads();

    // ---- Combine P0/P1 rows into s/t re/im; out[b,st,k,m,ri] ----
    for (int idx = tid; idx < 512; idx += 128) {
        const int kl = idx & 15;
        const int ks = (idx >> 4) & 1;
        const int b  = (idx >> 5) & 7;
        const int st = (idx >> 8) & 1;
        const int k  = kt2 * 32 + ks * 16 + kl;
        if (k >= NLAT) continue;
        const float* S0 = &ldsC[((ks * 2 + 0) * 16 + kl) * 32];
        const float* S1 = &ldsC[((ks * 2 + 1) * 16 + kl) * 32];
        float re, im;
        if (st == 0) {  // s: srl = P0[xr0] - P1[xi1]; sim = P0[xi0] + P1[xr1]
            re = S0[b]      - S1[24 + b];
            im = S0[8 + b]  + S1[16 + b];
        } else {        // t: trl = -P1[xi0] - P0[xr1]; tim = P1[xr0] - P0[xi1]
            re = -S1[8 + b] - S0[16 + b];
            im = S1[b]      - S0[24 + b];
        }
        size_t o = ((((size_t)b * 2 + st) * NLAT + k) * MMAX + m) * 2;
        float2 v = make_float2(re, im);
        *reinterpret_cast<float2*>(out + o) = v;
    }
}

extern "C" void kernel_launch(void* const* d_in, const int* in_sizes, int n_in,
                              void* d_out, int out_size, void* d_ws, size_t ws_size,
                              hipStream_t stream) {
    const float* x    = (const float*)d_in[0];
    const float* dpct = (const float*)d_in[1];
    float* out        = (float*)d_out;
    (void)in_sizes; (void)n_in; (void)out_size; (void)d_ws; (void)ws_size;

    dim3 grid(MMAX, 16);   // 257 m-values x 16 k-tile-pairs
    sht_wmma_kernel<<<grid, 128, 0, stream>>>(x, dpct, out);
}